// RetentiveAttention_76166950027414
// MI455X (gfx1250) — compile-verified
//
#include <hip/hip_runtime.h>
#include <hip/hip_bf16.h>

typedef __attribute__((ext_vector_type(2))) float        v2f;
typedef __attribute__((ext_vector_type(8))) float        v8f;
typedef __attribute__((ext_vector_type(4))) unsigned int v4u;
typedef __attribute__((ext_vector_type(8))) unsigned int v8u;

#define N_NODES   100000
#define N_EDGES   3200000
#define C_IN      128
#define C_OUT     128
#define BATCH     4
#define DECAY_F   0.7f
#define LN_EPS_F  1e-5f
#define XPAD      132          // padded row stride (floats): conflict-free b64 LDS reads

// LDS layout (floats). Total = 38016 floats = 152064 B (< 320 KB/WGP on gfx1250).
#define X_OFF     0                       // x tile: 128 rows x 132
#define WV_OFF    (128 * XPAD)            // W_v:    128 rows x 132
#define WK_OFF    (WV_OFF + 128 * XPAD)   // W_k:     16 rows x 132
#define WQ_OFF    (WK_OFF + 16 * XPAD)    // W_q:     16 rows x 132
#define LDS_FLOATS (WQ_OFF + 16 * XPAD)

// ---------------------------------------------------------------------------
// TDM 2-D tile load: global (row-major, row = 128 f32, stride 128) -> LDS with
// hardware padding of 4 DWORDs after every 128 DWORDs (=> 132-float LDS rows).
// D# per CDNA5 ISA ch.8: group0 = {count,lds_addr,global_addr,type=2},
// group1 = {data_size=4B, pad_enable, pad_interval=6 (128 DW), pad_amount=3
// (4 DW), tensor_dim0=128, tensor_dim1=rows, tile_dim0=128, tile_dim1=tile_rows,
// tensor_dim0_stride=128}. 2-D tensor -> VADDR2/3 omitted (NULL).
// All inputs must be wave-uniform (SGPR) values.
// ---------------------------------------------------------------------------
__device__ __forceinline__ void tdm_load_2d(unsigned lds_byte_off, const void* gptr,
                                            unsigned tensor_rows, unsigned tile_rows) {
  const unsigned long long ga = (unsigned long long)gptr;
  v4u g0;
  g0[0] = 1u;                                     // count=1, is_restore=0, gather=0
  g0[1] = lds_byte_off;                           // lds_addr (bytes)
  g0[2] = (unsigned)ga;                           // global_addr[31:0]
  g0[3] = (unsigned)(ga >> 32) | 0x80000000u;     // global_addr[56:32] | type=2<<30

  v8u g1;
  g1[0] = (2u << 16)        // data_size = 2 -> 4 bytes
        | (1u << 20)        // pad_enable
        | (6u << 22)        // pad_interval: 2^(6+1)=128 DWORDs between pads
        | (3u << 25);       // pad_amount: 3+1 = 4 DWORDs of padding
  g1[1] = (128u & 0xffffu) << 16;                           // tensor_dim0[15:0]
  g1[2] = ((tensor_rows & 0xffffu) << 16);                  // dim0[31:16]=0 | tensor_dim1[15:0]
  g1[3] = (tensor_rows >> 16) | (128u << 16);               // tensor_dim1[31:16] | tile_dim0
  g1[4] = tile_rows & 0xffffu;                              // tile_dim1 | tile_dim2=0
  g1[5] = 128u;                                             // tensor_dim0_stride[31:0]
  g1[6] = 0u;                                               // stride hi | tensor_dim1_stride lo
  g1[7] = 0u;

  asm volatile("tensor_load_to_lds %0, %1" :: "s"(g0), "s"(g1) : "memory");
}

// ---------------------------------------------------------------------------
// Kernel 1: fused K/Q/V projection with f32 WMMA (16x16x4).
// Block = 256 threads = 8 waves; block covers 128 nodes for batch b.
// Wave 0 issues 4 TDM descriptors (x tile + W_v + W_k + W_q -> LDS, padded),
// waits TENSORcnt, then the block barrier releases the compute.
// Each wave computes one 16-node tile:
//   values[16,128] -> d_out, weights0[16] = mean_k(key*query) -> wtot & wcur
// Fragment layouts per ISA 7.12.2 (32-bit, wave32):
//   A 16x4:  lane L: M=L&15, h=L>>4: v[0]=A[M][k0+2h], v[1]=A[M][k0+2h+1]
//   B 4x16:  lane L: N=L&15, h=L>>4: v[0]=W[N][k0+2h], v[1]=W[N][k0+2h+1]
//   D 16x16: lane L: N=L&15; vgpr r holds M=r+8h
// ---------------------------------------------------------------------------
__global__ __launch_bounds__(256)
void ra_proj_kernel(const float* __restrict__ x,
                    const float* __restrict__ Wk,
                    const float* __restrict__ Wq,
                    const float* __restrict__ Wv,
                    float* __restrict__ values,   // d_out, [B,N,128]
                    float* __restrict__ wtot,     // [N,4]
                    float* __restrict__ wcur) {   // [N,4]
  __shared__ float lds[LDS_FLOATS];

  const int lane = threadIdx.x & 31;
  const int wave = threadIdx.x >> 5;
  const int half = lane >> 4;     // 0 or 1
  const int lm   = lane & 15;     // M for A-frag, N for B/D-frags
  const int b    = blockIdx.y;
  const int blk_node0 = blockIdx.x * 128;
  const int node0     = blk_node0 + wave * 16;

  if (wave == 0) {
    const unsigned lbase = (unsigned)(unsigned long long)(const void*)lds;
    const unsigned rows_valid = (unsigned)min(128, N_NODES - blk_node0);
    tdm_load_2d(lbase + X_OFF  * 4u, x + ((size_t)b * N_NODES + blk_node0) * C_IN,
                rows_valid, 128u);
    tdm_load_2d(lbase + WV_OFF * 4u, Wv, 128u, 128u);
    tdm_load_2d(lbase + WK_OFF * 4u, Wk, 16u, 16u);
    tdm_load_2d(lbase + WQ_OFF * 4u, Wq, 16u, 16u);
    __builtin_amdgcn_s_wait_tensorcnt(0);
  }
  __syncthreads();
  if (node0 >= N_NODES) return;

  const v8f z8 = {0.f, 0.f, 0.f, 0.f, 0.f, 0.f, 0.f, 0.f};
  v8f accv[8];
#pragma unroll
  for (int i = 0; i < 8; ++i) accv[i] = z8;
  v8f acck = z8;
  v8f accq = z8;

  const int coff = 2 * half;
  const float* xrow = lds + X_OFF  + (wave * 16 + lm) * XPAD + coff;  // A frag base
  const float* wvp  = lds + WV_OFF + lm * XPAD + coff;                // B frag bases
  const float* wkp  = lds + WK_OFF + lm * XPAD + coff;
  const float* wqp  = lds + WQ_OFF + lm * XPAD + coff;

  for (int k0 = 0; k0 < C_IN; k0 += 4) {
    const v2f a  = *(const v2f*)(xrow + k0);
    const v2f bk = *(const v2f*)(wkp + k0);
    acck = __builtin_amdgcn_wmma_f32_16x16x4_f32(false, a, false, bk, (short)0, acck, false, false);
    const v2f bq = *(const v2f*)(wqp + k0);
    accq = __builtin_amdgcn_wmma_f32_16x16x4_f32(false, a, false, bq, (short)0, accq, false, false);
#pragma unroll
    for (int oc = 0; oc < 8; ++oc) {
      const v2f bv = *(const v2f*)(wvp + oc * 16 * XPAD + k0);
      accv[oc] = __builtin_amdgcn_wmma_f32_16x16x4_f32(false, a, false, bv, (short)0, accv[oc], false, false);
    }
  }

  // Store values: D[M][N], M=r+8h, col = oc*16+lm (halves cover 16 consecutive cols)
  float* outp = values + ((size_t)b * N_NODES + node0) * C_OUT;
#pragma unroll
  for (int oc = 0; oc < 8; ++oc) {
#pragma unroll
    for (int r = 0; r < 8; ++r) {
      outp[(size_t)(r + 8 * half) * C_OUT + oc * 16 + lm] = accv[oc][r];
    }
  }

  // weights0[M] = (1/16) * sum_kidx key[M][kidx]*query[M][kidx]
  // kidx = lm striped across 16 lanes of each half -> xor-shuffle reduce within half.
  float wsum = 0.f;
#pragma unroll
  for (int r = 0; r < 8; ++r) {
    float t = acck[r] * accq[r];
    t += __shfl_xor(t, 8, 32);
    t += __shfl_xor(t, 4, 32);
    t += __shfl_xor(t, 2, 32);
    t += __shfl_xor(t, 1, 32);
    if (lm == r) wsum = t * (1.0f / 16.0f);
  }
  if (lm < 8) {
    const int node = node0 + lm + 8 * half;   // lane(lm,half) owns M = lm + 8*half
    wtot[node * 4 + b] = wsum;
    wcur[node * 4 + b] = wsum;
  }
}

// ---------------------------------------------------------------------------
// Kernel 2a: zero a float buffer
// ---------------------------------------------------------------------------
__global__ __launch_bounds__(256)
void ra_zero_kernel(float* __restrict__ p, int n) {
  const int i = blockIdx.x * blockDim.x + threadIdx.x;
  if (i < n) p[i] = 0.f;
}

// ---------------------------------------------------------------------------
// Kernel 2b: SpMM scatter: nxt[row,:] += val*DECAY * cur[col,:]  ([N,4] layout)
// ---------------------------------------------------------------------------
__global__ __launch_bounds__(256)
void ra_spmm_kernel(const long long* __restrict__ rows,
                    const long long* __restrict__ cols,
                    const float* __restrict__ vals,
                    const float* __restrict__ cur,
                    float* __restrict__ nxt) {
  const int e = blockIdx.x * blockDim.x + threadIdx.x;
  if (e >= N_EDGES) return;
  const int r = (int)rows[e];
  const int c = (int)cols[e];
  const float v = vals[e] * DECAY_F;
  const float4 w = *(const float4*)(cur + (size_t)c * 4);
  float* dst = nxt + (size_t)r * 4;
  unsafeAtomicAdd(dst + 0, v * w.x);
  unsafeAtomicAdd(dst + 1, v * w.y);
  unsafeAtomicAdd(dst + 2, v * w.z);
  unsafeAtomicAdd(dst + 3, v * w.w);
}

// ---------------------------------------------------------------------------
// Kernel 2c: wtot += nxt
// ---------------------------------------------------------------------------
__global__ __launch_bounds__(256)
void ra_accum_kernel(float* __restrict__ wtot, const float* __restrict__ nxt, int n) {
  const int i = blockIdx.x * blockDim.x + threadIdx.x;
  if (i < n) wtot[i] += nxt[i];
}

// ---------------------------------------------------------------------------
// Kernel 3: vw = values*w, LayerNorm over C_OUT, in place in d_out.
// One wave per (b,n) row: float4 per lane, wave-wide shuffle reductions.
// ---------------------------------------------------------------------------
__global__ __launch_bounds__(256)
void ra_final_kernel(float* __restrict__ out,           // [B,N,128] (values in, out out)
                     const float* __restrict__ wtot,    // [N,4]
                     const float* __restrict__ gamma,
                     const float* __restrict__ beta) {
  const int lane = threadIdx.x & 31;
  const int row  = blockIdx.x * 8 + (threadIdx.x >> 5);   // row = b*N + n
  const int b = row / N_NODES;
  const int n = row - b * N_NODES;

  float* rp = out + (size_t)row * C_OUT + lane * 4;
  float4 v = *(const float4*)rp;
  const float w = wtot[(size_t)n * 4 + b];
  v.x *= w; v.y *= w; v.z *= w; v.w *= w;

  float s = v.x + v.y + v.z + v.w;
  s += __shfl_xor(s, 16, 32);
  s += __shfl_xor(s,  8, 32);
  s += __shfl_xor(s,  4, 32);
  s += __shfl_xor(s,  2, 32);
  s += __shfl_xor(s,  1, 32);
  const float mu = s * (1.0f / 128.0f);

  const float dx = v.x - mu, dy = v.y - mu, dz = v.z - mu, dw = v.w - mu;
  float s2 = dx * dx + dy * dy + dz * dz + dw * dw;
  s2 += __shfl_xor(s2, 16, 32);
  s2 += __shfl_xor(s2,  8, 32);
  s2 += __shfl_xor(s2,  4, 32);
  s2 += __shfl_xor(s2,  2, 32);
  s2 += __shfl_xor(s2,  1, 32);
  const float inv = rsqrtf(s2 * (1.0f / 128.0f) + LN_EPS_F);

  const float4 g  = *(const float4*)(gamma + lane * 4);
  const float4 be = *(const float4*)(beta  + lane * 4);
  float4 o;
  o.x = dx * inv * g.x + be.x;
  o.y = dy * inv * g.y + be.y;
  o.z = dz * inv * g.z + be.z;
  o.w = dw * inv * g.w + be.w;
  *(float4*)rp = o;
}

// ---------------------------------------------------------------------------
extern "C" void kernel_launch(void* const* d_in, const int* in_sizes, int n_in,
                              void* d_out, int out_size, void* d_ws, size_t ws_size,
                              hipStream_t stream) {
  const float*     x     = (const float*)d_in[0];
  const float*     Wk    = (const float*)d_in[1];
  const float*     Wq    = (const float*)d_in[2];
  const float*     Wv    = (const float*)d_in[3];
  const float*     gamma = (const float*)d_in[4];
  const float*     beta  = (const float*)d_in[5];
  const float*     ev    = (const float*)d_in[6];
  const long long* ei    = (const long long*)d_in[7];   // [2,E] int64

  float* out  = (float*)d_out;
  float* wtot = (float*)d_ws;                 // [N,4] accumulated weights
  float* wa   = wtot + (size_t)N_NODES * 4;   // [N,4] iter buffer A
  float* wb   = wa   + (size_t)N_NODES * 4;   // [N,4] iter buffer B

  const long long* rows = ei;
  const long long* cols = ei + N_EDGES;

  const int nw = N_NODES * 4;
  const int gw = (nw + 255) / 256;
  const int ge = (N_EDGES + 255) / 256;

  // 1) projections: values -> d_out, weights0 -> wtot & wa
  const int nblocks = (N_NODES + 127) / 128;        // 782 (last block partial)
  ra_proj_kernel<<<dim3(nblocks, BATCH), 256, 0, stream>>>(
      x, Wk, Wq, Wv, out, wtot, wa);

  // 2) iteration 1: wb = spmm(wa*DECAY); wtot += wb
  ra_zero_kernel<<<gw, 256, 0, stream>>>(wb, nw);
  ra_spmm_kernel<<<ge, 256, 0, stream>>>(rows, cols, ev, wa, wb);
  ra_accum_kernel<<<gw, 256, 0, stream>>>(wtot, wb, nw);

  // 3) iteration 2: wa = spmm(wb*DECAY); wtot += wa
  ra_zero_kernel<<<gw, 256, 0, stream>>>(wa, nw);
  ra_spmm_kernel<<<ge, 256, 0, stream>>>(rows, cols, ev, wb, wa);
  ra_accum_kernel<<<gw, 256, 0, stream>>>(wtot, wa, nw);

  // 4) scale + layernorm, in place in d_out (8 rows per 256-thread block)
  ra_final_kernel<<<(BATCH * N_NODES) / 8, 256, 0, stream>>>(out, wtot, gamma, beta);
}